// GraphConvolution_34557306864322
// MI455X (gfx1250) — compile-verified
//
#include <hip/hip_runtime.h>

typedef __attribute__((ext_vector_type(2))) float v2f;
typedef __attribute__((ext_vector_type(8))) float v8f;

#define C_DIM 128

// ---------------- degree kernels ----------------
__global__ void k_deg_init(float* __restrict__ deg, int N) {
    int i = blockIdx.x * blockDim.x + threadIdx.x;
    if (i < N) deg[i] = 1.0f;  // self loop contributes 1 to each degree
}

__global__ void k_deg_accum(const int* __restrict__ row, float* __restrict__ deg, int E) {
    int e = blockIdx.x * blockDim.x + threadIdx.x;
    if (e < E) atomicAdd(&deg[row[e]], 1.0f);
}

__global__ void k_deg_rsqrt(float* __restrict__ deg, int N) {
    int i = blockIdx.x * blockDim.x + threadIdx.x;
    if (i < N) {
        float d = deg[i];
        deg[i] = (d > 0.0f) ? rsqrtf(d) : 0.0f;
    }
}

// ---------------- WMMA GEMM: support = x @ W^T + b ----------------
// Block = 256 threads = 8 waves. Each wave owns a 16-row strip and the full
// 128 output channels (8 tiles of 16x16, K-loop of 32 x V_WMMA_F32_16X16X4_F32).
// A layout (f32 16x4): lane = 16*kh + m ; VGPR0/1 = x[m][k0+2*kh +0/1]
// B layout (f32 4x16): lane = 16*kh + n ; VGPR0/1 = W[n][k0+2*kh +0/1] (B = W^T)
// C/D layout: lane = 16*half + n ; VGPR v = row (v + 8*half)
__global__ __launch_bounds__(256) void k_gemm_wmma(const float* __restrict__ x,
                                                   const float* __restrict__ W,
                                                   const float* __restrict__ bias,
                                                   float* __restrict__ support,
                                                   int N) {
    __shared__ float sW[C_DIM * C_DIM];  // 64 KB staged weights

    // cooperative 128-bit staging of W into LDS (uniform trip count -> EXEC stays full)
    const float4* Wv = (const float4*)W;
    float4* sWv = (float4*)sW;
    #pragma unroll 4
    for (int i = threadIdx.x; i < (C_DIM * C_DIM) / 4; i += 256) sWv[i] = Wv[i];
    __syncthreads();

    const int lane = threadIdx.x & 31;
    const int wave = threadIdx.x >> 5;
    const int r    = lane & 15;   // M index for A, N index for B/C
    const int kh   = lane >> 4;   // K-half selector (0: k,k+1 ; 1: k+2,k+3)
    const int rowBase = blockIdx.x * 128 + wave * 16;

    // init accumulators with bias (bias depends only on output channel = lane%16)
    v8f acc[8];
    #pragma unroll
    for (int j = 0; j < 8; ++j) {
        float bv = bias[j * 16 + r];
        #pragma unroll
        for (int v = 0; v < 8; ++v) acc[j][v] = bv;
    }

    int arow = rowBase + r;
    if (arow > N - 1) arow = N - 1;  // tail: read valid row, stores are guarded
    const float* xrow = x + (size_t)arow * C_DIM;

    for (int k = 0; k < C_DIM; k += 4) {
        v2f a;
        a.x = xrow[k + 2 * kh];
        a.y = xrow[k + 2 * kh + 1];
        #pragma unroll
        for (int j = 0; j < 8; ++j) {
            v2f bf;
            bf.x = sW[(j * 16 + r) * C_DIM + k + 2 * kh];
            bf.y = sW[(j * 16 + r) * C_DIM + k + 2 * kh + 1];
            acc[j] = __builtin_amdgcn_wmma_f32_16x16x4_f32(
                /*neg_a=*/false, a, /*neg_b=*/false, bf,
                /*c_mod=*/(short)0, acc[j],
                /*reuse_a=*/false, /*reuse_b=*/false);
        }
    }

    #pragma unroll
    for (int v = 0; v < 8; ++v) {
        int m = rowBase + v + 8 * kh;
        if (m < N) {
            float* drow = support + (size_t)m * C_DIM;
            #pragma unroll
            for (int j = 0; j < 8; ++j) drow[j * 16 + r] = acc[j][v];
        }
    }
}

// ---------------- self-loop term: out[i] = dinv[i]^2 * support[i] ----------------
__global__ void k_self(const float* __restrict__ support, const float* __restrict__ dinv,
                       float* __restrict__ out, int N) {
    int t = blockIdx.x * blockDim.x + threadIdx.x;  // one float4 per thread
    int total = N * (C_DIM / 4);
    if (t < total) {
        int i = t / (C_DIM / 4);
        float s = dinv[i];
        s = s * s;
        float4 v = ((const float4*)support)[t];
        float4 o = make_float4(v.x * s, v.y * s, v.z * s, v.w * s);
        ((float4*)out)[t] = o;
    }
}

// ---------------- edge scatter: out[r] += dinv[r]*dinv[c]*support[c] ----------------
// One wave per edge; each lane handles 4 channels via float4 gather + 4 f32 atomics.
__global__ __launch_bounds__(256) void k_scatter(const int* __restrict__ row,
                                                 const int* __restrict__ col,
                                                 const float* __restrict__ dinv,
                                                 const float* __restrict__ support,
                                                 float* __restrict__ out, int E) {
    long long gtid = (long long)blockIdx.x * blockDim.x + threadIdx.x;
    int e    = (int)(gtid >> 5);
    int lane = (int)(gtid & 31);
    if (e < E) {
        int r = row[e];
        int c = col[e];
        float norm = dinv[r] * dinv[c];
        const float4 sv = *(const float4*)(support + (size_t)c * C_DIM + lane * 4);
        float* orow = out + (size_t)r * C_DIM + lane * 4;
        atomicAdd(orow + 0, norm * sv.x);
        atomicAdd(orow + 1, norm * sv.y);
        atomicAdd(orow + 2, norm * sv.z);
        atomicAdd(orow + 3, norm * sv.w);
    }
}

extern "C" void kernel_launch(void* const* d_in, const int* in_sizes, int n_in,
                              void* d_out, int out_size, void* d_ws, size_t ws_size,
                              hipStream_t stream) {
    const float* x  = (const float*)d_in[0];
    const int*   ei = (const int*)d_in[1];   // [2, E]
    const float* W  = (const float*)d_in[2]; // [C_out, C_in]
    const float* b  = (const float*)d_in[3]; // [C_out]
    float* out = (float*)d_out;

    const int N = in_sizes[0] / C_DIM;
    const int E = in_sizes[1] / 2;
    const int* row = ei;
    const int* col = ei + E;

    // workspace layout: [deg/dinv: N floats (padded)] [support: N*128 floats]
    float* deg = (float*)d_ws;
    size_t degPad = ((size_t)N + 255) & ~(size_t)255;
    float* support = deg + degPad;

    k_deg_init<<<(N + 255) / 256, 256, 0, stream>>>(deg, N);
    k_deg_accum<<<(E + 255) / 256, 256, 0, stream>>>(row, deg, E);
    k_deg_rsqrt<<<(N + 255) / 256, 256, 0, stream>>>(deg, N);

    k_gemm_wmma<<<(N + 127) / 128, 256, 0, stream>>>(x, W, b, support, N);

    int totalVec4 = N * (C_DIM / 4);
    k_self<<<(totalVec4 + 255) / 256, 256, 0, stream>>>(support, deg, out, N);

    long long sthreads = (long long)E * 32;
    int sblocks = (int)((sthreads + 255) / 256);
    k_scatter<<<sblocks, 256, 0, stream>>>(row, col, deg, support, out, E);
}